// ClimbingRouteGNN_20590073217288
// MI455X (gfx1250) — compile-verified
//
#include <hip/hip_runtime.h>
#include <hip/hip_bf16.h>

// ---------------- problem constants (from reference) ----------------
#define BGRAPH   128
#define NNODE    512
#define NTOT     65536      // BGRAPH * NNODE
#define NEDGE    524288     // NTOT * 8
#define INDIM    128
#define H1DIM    256
#define H2DIM    128
#define KPOOL    256

typedef float v2f __attribute__((ext_vector_type(2)));
typedef float v8f __attribute__((ext_vector_type(8)));

// --------- monotonic float<->uint encoding for atomic max on floats ---------
__device__ __forceinline__ unsigned encF(float v) {
    unsigned b = __float_as_uint(v);
    return (b & 0x80000000u) ? ~b : (b | 0x80000000u);
}
__device__ __forceinline__ float decF(unsigned e) {
    unsigned b = (e & 0x80000000u) ? (e & 0x7FFFFFFFu) : ~e;
    return __uint_as_float(b);
}
__device__ __forceinline__ float lrelu02(float x) { return x > 0.0f ? x : 0.2f * x; }

// ---------------- k1: degree = segment_sum(1, row) ----------------
__global__ void k_degree(const int* __restrict__ row, float* __restrict__ deg, int E) {
    int e = blockIdx.x * 256 + threadIdx.x;
    if (e < E) atomicAdd(&deg[row[e]], 1.0f);
}

// ---------------- k2: dinv ----------------
__global__ void k_dinv(const float* __restrict__ deg, float* __restrict__ dinv, int n) {
    int i = blockIdx.x * 256 + threadIdx.x;
    if (i < n) {
        float d = deg[i];
        dinv[i] = (d > 0.0f) ? rsqrtf(fmaxf(d, 1.0f)) : 0.0f;
    }
}

// ---------------- k3: tx1[col] += w_e * x[row]   (wave per edge, 4 feats/lane) ----------------
__global__ void k_cheb_scatter(const float* __restrict__ x,
                               const int* __restrict__ row, const int* __restrict__ col,
                               const float* __restrict__ dinv, float* __restrict__ tx1, int E) {
    int item = blockIdx.x * 8 + (threadIdx.x >> 5);
    if (item >= E) return;
    int lane = threadIdx.x & 31;
    int r = row[item], c = col[item];
    float w = -dinv[r] * dinv[c];
    const float* xs = x   + (size_t)r * INDIM;
    float*       td = tx1 + (size_t)c * INDIM;
#pragma unroll
    for (int f = lane; f < INDIM; f += 32)
        atomicAdd(&td[f], w * xs[f]);
}

// ---------------- WMMA fp32 GEMM: C = act( A0@B0 [+ A1@B1] + bias ) ----------------
// A: [M,K] row-major, B: [K,N] row-major, C: [M,N].  K, N compile-time.
// Block = 256 threads = 8 waves; wave computes a 16x16 tile; block covers 128 rows x 16 cols.
// The K x 16 strip of B (and B1 when DUAL) is staged in LDS once per block: all 8 waves
// share nBase, so this removes 8x-redundant strided global loads and the per-step 64-bit
// address arithmetic the compiler was emitting.
template <int K, int N, bool DUAL, bool RELU, bool BIAS>
__global__ __launch_bounds__(256) void k_wmma_gemm(const float* __restrict__ A0,
                                                   const float* __restrict__ B0,
                                                   const float* __restrict__ A1,
                                                   const float* __restrict__ B1,
                                                   const float* __restrict__ bias,
                                                   float* __restrict__ C) {
    __shared__ float Bs[(DUAL ? 2 : 1) * K * 16];

    const int t     = threadIdx.x;
    const int nBase = blockIdx.x * 16;

    // cooperative stage of B strip(s): Bs[k][n] = B[k][nBase+n]
#pragma unroll
    for (int idx = t; idx < K * 16; idx += 256)
        Bs[idx] = B0[(size_t)(idx >> 4) * N + nBase + (idx & 15)];
    if (DUAL) {
#pragma unroll
        for (int idx = t; idx < K * 16; idx += 256)
            Bs[K * 16 + idx] = B1[(size_t)(idx >> 4) * N + nBase + (idx & 15)];
    }
    __syncthreads();

    const int lane  = t & 31;
    const int wave  = t >> 5;
    const int mBase = blockIdx.y * 128 + wave * 16;
    const int half  = lane >> 4;   // 0: lanes 0-15 (K=k0,k0+1), 1: lanes 16-31 (K=k0+2,k0+3)
    const int l16   = lane & 15;

    v8f acc = {};
    // A fragment: row (mBase+l16), contiguous pair at k0 + 2*half  -> aligned float2 loads
    const v2f* aP0 = (const v2f*)(A0 + (size_t)(mBase + l16) * K) + half;
#pragma unroll
    for (int k0 = 0; k0 < K; k0 += 4) {
        v2f a = aP0[k0 >> 1];
        v2f b;
        b.x = Bs[(k0 + 2 * half + 0) * 16 + l16];
        b.y = Bs[(k0 + 2 * half + 1) * 16 + l16];
        acc = __builtin_amdgcn_wmma_f32_16x16x4_f32(false, a, false, b,
                                                    (short)0, acc, false, false);
    }
    if (DUAL) {
        const v2f* aP1 = (const v2f*)(A1 + (size_t)(mBase + l16) * K) + half;
#pragma unroll
        for (int k0 = 0; k0 < K; k0 += 4) {
            v2f a = aP1[k0 >> 1];
            v2f b;
            b.x = Bs[K * 16 + (k0 + 2 * half + 0) * 16 + l16];
            b.y = Bs[K * 16 + (k0 + 2 * half + 1) * 16 + l16];
            acc = __builtin_amdgcn_wmma_f32_16x16x4_f32(false, a, false, b,
                                                        (short)0, acc, false, false);
        }
    }

    float bv = BIAS ? bias[nBase + l16] : 0.0f;
#pragma unroll
    for (int r = 0; r < 8; ++r) {
        int rowIdx = mBase + r + 8 * half;   // VGPR r: M=r (lanes 0-15), M=r+8 (lanes 16-31)
        float v = acc[r] + bv;
        if (RELU) v = fmaxf(v, 0.0f);
        C[(size_t)rowIdx * N + nBase + l16] = v;
    }
}

// ---------------- k5: per-node attention coefficients (wave per node) ----------------
__global__ void k_att_node(const float* __restrict__ z, const float* __restrict__ asrc,
                           const float* __restrict__ adst, float* __restrict__ a_s,
                           float* __restrict__ a_d, int n) {
    int node = blockIdx.x * 8 + (threadIdx.x >> 5);
    if (node >= n) return;
    int lane = threadIdx.x & 31;
    const float* zr = z + (size_t)node * H2DIM;
    float s = 0.0f, d = 0.0f;
#pragma unroll
    for (int f = lane; f < H2DIM; f += 32) {
        float v = zr[f];
        s += v * asrc[f];
        d += v * adst[f];
    }
#pragma unroll
    for (int off = 16; off > 0; off >>= 1) {
        s += __shfl_xor(s, off);
        d += __shfl_xor(d, off);
    }
    if (lane == 0) { a_s[node] = s; a_d[node] = d; }
}

// ---------------- k6: segment max of leaky_relu(a_s[r]+a_d[c]) over col (+self loops) ------------
__global__ void k_seg_max(const int* __restrict__ row, const int* __restrict__ col,
                          const float* __restrict__ a_s, const float* __restrict__ a_d,
                          unsigned* __restrict__ m_enc, int E, int n) {
    int t = blockIdx.x * 256 + threadIdx.x;
    if (t >= E + n) return;
    int r, c;
    if (t < E) { r = row[t]; c = col[t]; } else { r = c = t - E; }
    float e = lrelu02(a_s[r] + a_d[c]);
    atomicMax(&m_enc[c], encF(e));
}

// ---------------- k7: segment sum of exp(e - m[c]) ----------------
__global__ void k_seg_sum(const int* __restrict__ row, const int* __restrict__ col,
                          const float* __restrict__ a_s, const float* __restrict__ a_d,
                          const unsigned* __restrict__ m_enc, float* __restrict__ ssum,
                          int E, int n) {
    int t = blockIdx.x * 256 + threadIdx.x;
    if (t >= E + n) return;
    int r, c;
    if (t < E) { r = row[t]; c = col[t]; } else { r = c = t - E; }
    float e = lrelu02(a_s[r] + a_d[c]);
    atomicAdd(&ssum[c], expf(e - decF(m_enc[c])));
}

// ---------------- k8: g[c] += alpha * z[r]  (wave per edge item) ----------------
__global__ void k_gat_scatter(const int* __restrict__ row, const int* __restrict__ col,
                              const float* __restrict__ a_s, const float* __restrict__ a_d,
                              const unsigned* __restrict__ m_enc, const float* __restrict__ ssum,
                              const float* __restrict__ z, float* __restrict__ g, int E, int n) {
    int item = blockIdx.x * 8 + (threadIdx.x >> 5);
    if (item >= E + n) return;
    int lane = threadIdx.x & 31;
    int r, c;
    if (item < E) { r = row[item]; c = col[item]; } else { r = c = item - E; }
    float e = lrelu02(a_s[r] + a_d[c]);
    float w = expf(e - decF(m_enc[c])) / ssum[c];
    const float* zr = z + (size_t)r * H2DIM;
    float*       gd = g + (size_t)c * H2DIM;
#pragma unroll
    for (int f = lane; f < H2DIM; f += 32)
        atomicAdd(&gd[f], w * zr[f]);
}

// ---------------- k9: 1/||pool_p|| ----------------
__global__ void k_pnorm(const float* __restrict__ p, float* __restrict__ inv_norm) {
    __shared__ float red[128];
    int t = threadIdx.x;
    float v = p[t];
    red[t] = v * v;
    __syncthreads();
    for (int s = 64; s > 0; s >>= 1) {
        if (t < s) red[t] += red[t + s];
        __syncthreads();
    }
    if (t == 0) inv_norm[0] = rsqrtf(red[0]);
}

// ---------------- k10: g = relu(g + b); score = tanh(g.p / ||p||)  (wave per node) -------------
__global__ void k_node_final(float* __restrict__ g, const float* __restrict__ gb,
                             const float* __restrict__ p, const float* __restrict__ inv_norm,
                             float* __restrict__ score, int n) {
    int node = blockIdx.x * 8 + (threadIdx.x >> 5);
    if (node >= n) return;
    int lane = threadIdx.x & 31;
    float* gr = g + (size_t)node * H2DIM;
    float dot = 0.0f;
#pragma unroll
    for (int f = lane; f < H2DIM; f += 32) {
        float v = fmaxf(gr[f] + gb[f], 0.0f);
        gr[f] = v;
        dot += v * p[f];
    }
#pragma unroll
    for (int off = 16; off > 0; off >>= 1) dot += __shfl_xor(dot, off);
    if (lane == 0) score[node] = tanhf(dot * inv_norm[0]);
}

// ---------------- k11: per-graph top-k (rank counting) + gated mean pool ----------------
__global__ void k_topk_mean(const float* __restrict__ g, const float* __restrict__ score,
                            float* __restrict__ gm) {
    __shared__ float s_sc[NNODE];
    __shared__ float s_acc[H2DIM];
    int b = blockIdx.x, t = threadIdx.x;
    s_sc[t]       = score[(size_t)b * NNODE + t];
    s_sc[t + 256] = score[(size_t)b * NNODE + t + 256];
    if (t < H2DIM) s_acc[t] = 0.0f;
    __syncthreads();
#pragma unroll
    for (int jj = 0; jj < 2; ++jj) {
        int j = t + jj * 256;
        float sj = s_sc[j];
        int rank = 0;
        for (int i = 0; i < NNODE; ++i) {
            float si = s_sc[i];
            rank += (si > sj) || (si == sj && i < j);   // lax.top_k tie-break: lower index first
        }
        if (rank < KPOOL) {
            const float* gr = g + ((size_t)b * NNODE + j) * H2DIM;
            for (int f = 0; f < H2DIM; ++f)
                atomicAdd(&s_acc[f], sj * gr[f]);
        }
    }
    __syncthreads();
    if (t < H2DIM) gm[(size_t)b * H2DIM + t] = s_acc[t] * (1.0f / (float)KPOOL);
}

// ---------------- k12: MLP head, one block (256 thr) per graph ----------------
__global__ void k_head(const float* __restrict__ gm, const float* __restrict__ w1,
                       const float* __restrict__ b1, const float* __restrict__ w2,
                       const float* __restrict__ b2, float* __restrict__ out) {
    __shared__ float s_g[H2DIM];
    __shared__ float s_red[256];
    int b = blockIdx.x, t = threadIdx.x;
    if (t < H2DIM) s_g[t] = gm[(size_t)b * H2DIM + t];
    __syncthreads();
    float h = 0.0f;
    for (int i = 0; i < H2DIM; ++i) h += s_g[i] * w1[(size_t)i * H1DIM + t];
    h = fmaxf(h + b1[t], 0.0f);
    s_red[t] = h * w2[t];
    __syncthreads();
    for (int s = 128; s > 0; s >>= 1) {
        if (t < s) s_red[t] += s_red[t + s];
        __syncthreads();
    }
    if (t == 0) out[b] = s_red[0] + b2[0];
}

// =====================================================================
extern "C" void kernel_launch(void* const* d_in, const int* in_sizes, int n_in,
                              void* d_out, int out_size, void* d_ws, size_t ws_size,
                              hipStream_t stream) {
    const float* x       = (const float*)d_in[0];
    const int*   eidx    = (const int*)d_in[1];
    const float* cheb_w0 = (const float*)d_in[3];
    const float* cheb_w1 = (const float*)d_in[4];
    const float* cheb_b  = (const float*)d_in[5];
    const float* gat_w   = (const float*)d_in[6];
    const float* gat_as  = (const float*)d_in[7];
    const float* gat_ad  = (const float*)d_in[8];
    const float* gat_b   = (const float*)d_in[9];
    const float* pool_p  = (const float*)d_in[10];
    const float* lin1_w  = (const float*)d_in[11];
    const float* lin1_b  = (const float*)d_in[12];
    const float* lin2_w  = (const float*)d_in[13];
    const float* lin2_b  = (const float*)d_in[14];
    float* out = (float*)d_out;

    const int* row = eidx;
    const int* col = eidx + NEDGE;

    // ------- workspace layout (floats) -------
    float* ws = (float*)d_ws;
    float*    deg      = ws;                    // 65536
    float*    dinv     = ws + 65536;            // 65536
    float*    a_s      = ws + 131072;           // 65536
    float*    a_d      = ws + 196608;           // 65536
    float*    ssum     = ws + 262144;           // 65536
    float*    score    = ws + 327680;           // 65536
    float*    inv_norm = ws + 393216;           // 1
    float*    gm       = ws + 393280;           // 16384
    unsigned* m_enc    = (unsigned*)(ws + 409664); // 65536
    float*    tx1z     = ws + 524288;           // 8388608  (tx1, later reused as z)
    float*    h        = ws + 8912896;          // 16777216
    float*    g        = ws + 25690112;         // 8388608

    // ------- zero init (graph-capturable memsets) -------
    hipMemsetAsync(deg,   0, NTOT * sizeof(float), stream);
    hipMemsetAsync(ssum,  0, NTOT * sizeof(float), stream);
    hipMemsetAsync(m_enc, 0, NTOT * sizeof(unsigned), stream); // 0 encodes < any real value
    hipMemsetAsync(tx1z,  0, (size_t)NTOT * INDIM * sizeof(float), stream);
    hipMemsetAsync(g,     0, (size_t)NTOT * H2DIM * sizeof(float), stream);

    // ------- ChebConv -------
    k_degree<<<NEDGE / 256, 256, 0, stream>>>(row, deg, NEDGE);
    k_dinv<<<NTOT / 256, 256, 0, stream>>>(deg, dinv, NTOT);
    k_cheb_scatter<<<NEDGE / 8, 256, 0, stream>>>(x, row, col, dinv, tx1z, NEDGE);
    // h = relu(x@W0 + tx1@W1 + b)  [65536 x 256]
    {
        dim3 grid(H1DIM / 16, NTOT / 128);
        k_wmma_gemm<INDIM, H1DIM, true, true, true>
            <<<grid, 256, 0, stream>>>(x, cheb_w0, tx1z, cheb_w1, cheb_b, h);
    }

    // ------- GATConv -------
    // z = h @ gat_w   [65536 x 128]  (reuse tx1 buffer as z)
    {
        dim3 grid(H2DIM / 16, NTOT / 128);
        k_wmma_gemm<H1DIM, H2DIM, false, false, false>
            <<<grid, 256, 0, stream>>>(h, gat_w, nullptr, nullptr, nullptr, tx1z);
    }
    float* z = tx1z;
    k_att_node<<<NTOT / 8, 256, 0, stream>>>(z, gat_as, gat_ad, a_s, a_d, NTOT);
    {
        int items = NEDGE + NTOT;
        int grid1 = (items + 255) / 256;
        k_seg_max<<<grid1, 256, 0, stream>>>(row, col, a_s, a_d, m_enc, NEDGE, NTOT);
        k_seg_sum<<<grid1, 256, 0, stream>>>(row, col, a_s, a_d, m_enc, ssum, NEDGE, NTOT);
        k_gat_scatter<<<(items + 7) / 8, 256, 0, stream>>>(row, col, a_s, a_d, m_enc, ssum,
                                                           z, g, NEDGE, NTOT);
    }

    // ------- TopK pooling + mean + head -------
    k_pnorm<<<1, 128, 0, stream>>>(pool_p, inv_norm);
    k_node_final<<<NTOT / 8, 256, 0, stream>>>(g, gat_b, pool_p, inv_norm, score, NTOT);
    k_topk_mean<<<BGRAPH, 256, 0, stream>>>(g, score, gm);
    k_head<<<BGRAPH, 256, 0, stream>>>(gm, lin1_w, lin1_b, lin2_w, lin2_b, out);
}